// DKD_56212531970517
// MI455X (gfx1250) — compile-verified
//
#include <hip/hip_runtime.h>
#include <hip/hip_bf16.h>
#include <math.h>

typedef __attribute__((ext_vector_type(2))) float v2f;
typedef __attribute__((ext_vector_type(8))) float v8f;

#define DKD_B 8
#define DKD_H 1024
#define DKD_W 1024
#define DKD_N (DKD_H * DKD_W)          // 1048576
#define DKD_TOT (DKD_B * DKD_N)        // 8388608
#define TOPK 4096
#define NBINS 65536
#define CAND_CAP 8192
#define SORT_N 8192

// ---------------- workspace layout (bytes) ----------------
#define OFF_A      0ull                       // fp32 rowmax scratch: 33554432
#define OFF_MASK   33554432ull                // u8 mask:             8388608
#define OFF_TMPB   41943040ull                // u8 row-or scratch:   8388608
#define OFF_SUPP   50331648ull                // u8 supp mask:        8388608
#define OFF_HIST   58720256ull                // u32 hist 8*65536:    2097152
#define OFF_THR    60817408ull                // u32 [8]
#define OFF_CNT    60817440ull                // u32 [8]
#define OFF_CAND   60817472ull                // u64 8*8192:          524288
#define OFF_KIDX   61341760ull                // i32 8*4096:          131072
// total ~58.6 MB

// ---------------- NMS maxpool passes ----------------

// A[p] = rowmax5( mode ? (supp?0:s) : s )
__global__ void dkd_rowmax(const float* __restrict__ s,
                           const unsigned char* __restrict__ suppb,
                           float* __restrict__ A, int mode) {
  int gid = blockIdx.x * 256 + threadIdx.x;
  if (gid >= DKD_TOT) return;
  int x = gid & (DKD_W - 1);
  int row = gid - x;
  // next-row prefetch: working set is L2-resident (192MB), pull line early
  if (gid + DKD_W < DKD_TOT) __builtin_prefetch(&s[gid + DKD_W], 0, 2);
  int x0 = x - 2; if (x0 < 0) x0 = 0;
  int x1 = x + 2; if (x1 > DKD_W - 1) x1 = DKD_W - 1;
  float m = -1e30f;
  for (int xx = x0; xx <= x1; ++xx) {
    float v = s[row + xx];
    if (mode && suppb[row + xx]) v = 0.0f;
    m = fmaxf(m, v);
  }
  A[gid] = m;
}

// mode 0: mask = (s == colmax5(A));  mode 1: mask |= (!supp && s == colmax5(A))
__global__ void dkd_colcmp(const float* __restrict__ s,
                           const float* __restrict__ A,
                           const unsigned char* __restrict__ suppb,
                           unsigned char* __restrict__ maskb, int mode) {
  int gid = blockIdx.x * 256 + threadIdx.x;
  if (gid >= DKD_TOT) return;
  int x = gid & (DKD_W - 1);
  int y = (gid >> 10) & (DKD_H - 1);
  int b = gid >> 20;
  int y0 = y - 2; if (y0 < 0) y0 = 0;
  int y1 = y + 2; if (y1 > DKD_H - 1) y1 = DKD_H - 1;
  int base = b * DKD_N + x;
  float m = -1e30f;
  for (int yy = y0; yy <= y1; ++yy) m = fmaxf(m, A[base + yy * DKD_W]);
  float sv = s[gid];
  if (mode == 0) {
    maskb[gid] = (sv == m) ? 1 : 0;
  } else {
    unsigned char sp = suppb[gid];
    float ss = sp ? 0.0f : sv;
    if (!sp && ss == m) maskb[gid] = 1;
  }
}

__global__ void dkd_rowor(const unsigned char* __restrict__ maskb,
                          unsigned char* __restrict__ tmpb) {
  int gid = blockIdx.x * 256 + threadIdx.x;
  if (gid >= DKD_TOT) return;
  int x = gid & (DKD_W - 1);
  int row = gid - x;
  int x0 = x - 2; if (x0 < 0) x0 = 0;
  int x1 = x + 2; if (x1 > DKD_W - 1) x1 = DKD_W - 1;
  unsigned char m = 0;
  for (int xx = x0; xx <= x1; ++xx) m |= maskb[row + xx];
  tmpb[gid] = m;
}

__global__ void dkd_color(const unsigned char* __restrict__ tmpb,
                          unsigned char* __restrict__ suppb) {
  int gid = blockIdx.x * 256 + threadIdx.x;
  if (gid >= DKD_TOT) return;
  int x = gid & (DKD_W - 1);
  int y = (gid >> 10) & (DKD_H - 1);
  int b = gid >> 20;
  int y0 = y - 2; if (y0 < 0) y0 = 0;
  int y1 = y + 2; if (y1 > DKD_H - 1) y1 = DKD_H - 1;
  int base = b * DKD_N + x;
  unsigned char m = 0;
  for (int yy = y0; yy <= y1; ++yy) m |= tmpb[base + yy * DKD_W];
  suppb[gid] = m;
}

// ---------------- top-k select ----------------

__global__ void dkd_zero(unsigned* __restrict__ hist, unsigned* __restrict__ cnt) {
  int i = blockIdx.x * 256 + threadIdx.x;
  if (i < DKD_B * NBINS) hist[i] = 0u;
  else if (i < DKD_B * NBINS + DKD_B) cnt[i - DKD_B * NBINS] = 0u;
}

__device__ __forceinline__ int dkd_bin(float v) {
  int bin = (int)(v * 65536.0f);
  if (bin < 0) bin = 0;
  if (bin > NBINS - 1) bin = NBINS - 1;
  return bin;
}

__global__ void dkd_hist(const float* __restrict__ s,
                         const unsigned char* __restrict__ maskb,
                         unsigned* __restrict__ hist) {
  int gid = blockIdx.x * 256 + threadIdx.x;
  if (gid >= DKD_TOT) return;
  int x = gid & (DKD_W - 1);
  int y = (gid >> 10) & (DKD_H - 1);
  int b = gid >> 20;
  bool inner = (x >= 2) && (x < DKD_W - 2) && (y >= 2) && (y < DKD_H - 2);
  if (inner && maskb[gid]) {
    atomicAdd(&hist[b * NBINS + dkd_bin(s[gid])], 1u);
  }
}

// one block / batch; find max T with count(bins >= T) >= TOPK
__global__ void dkd_thresh(const unsigned* __restrict__ hist,
                           unsigned* __restrict__ thr) {
  __shared__ unsigned suf[1024];
  int b = blockIdx.x;
  int tid = threadIdx.x;
  const unsigned* hb = hist + b * NBINS;
  unsigned sum = 0;
  int base = tid * 64;
  for (int q = 0; q < 64; ++q) sum += hb[base + q];
  suf[tid] = sum;
  __syncthreads();
  for (int off = 1; off < 1024; off <<= 1) {
    unsigned v = (tid + off < 1024) ? suf[tid + off] : 0u;
    __syncthreads();
    suf[tid] += v;
    __syncthreads();
  }
  unsigned Sc = suf[tid];
  unsigned Sn = (tid < 1023) ? suf[tid + 1] : 0u;
  if (tid == 0 && Sc < TOPK) thr[b] = 0u;
  if (Sc >= TOPK && Sn < TOPK) {
    unsigned running = Sn;
    unsigned T = (unsigned)base;
    for (int q = 63; q >= 0; --q) {
      running += hb[base + q];
      if (running >= TOPK) { T = (unsigned)(base + q); break; }
    }
    thr[b] = T;
  }
}

__global__ void dkd_compact(const float* __restrict__ s,
                            const unsigned char* __restrict__ maskb,
                            const unsigned* __restrict__ thr,
                            unsigned* __restrict__ cnt,
                            unsigned long long* __restrict__ cand) {
  int gid = blockIdx.x * 256 + threadIdx.x;
  if (gid >= DKD_TOT) return;
  int x = gid & (DKD_W - 1);
  int y = (gid >> 10) & (DKD_H - 1);
  int b = gid >> 20;
  bool inner = (x >= 2) && (x < DKD_W - 2) && (y >= 2) && (y < DKD_H - 2);
  if (!(inner && maskb[gid])) return;
  float v = s[gid];
  if ((unsigned)dkd_bin(v) < thr[b]) return;
  unsigned pos = atomicAdd(&cnt[b], 1u);
  if (pos < CAND_CAP) {
    unsigned idx = (unsigned)(gid & (DKD_N - 1));
    cand[b * CAND_CAP + pos] =
        ((unsigned long long)__float_as_uint(v) << 32) | (unsigned)(~idx);
  }
}

// one block / batch: bitonic sort SORT_N keys descending, emit top 4096 indices
__global__ void dkd_sort(const unsigned long long* __restrict__ cand,
                         const unsigned* __restrict__ cnt,
                         int* __restrict__ kpidx) {
  extern __shared__ unsigned long long keys[];
  int b = blockIdx.x;
  int tid = threadIdx.x;
  unsigned c = cnt[b];
  if (c > CAND_CAP) c = CAND_CAP;
  for (int i = tid; i < SORT_N; i += 1024)
    keys[i] = (i < (int)c) ? cand[b * CAND_CAP + i] : 0ull;
  __syncthreads();
  for (int k = 2; k <= SORT_N; k <<= 1) {
    for (int j = k >> 1; j > 0; j >>= 1) {
      for (int i = tid; i < SORT_N; i += 1024) {
        int ixj = i ^ j;
        if (ixj > i) {
          bool desc = ((i & k) == 0);
          unsigned long long a = keys[i], d = keys[ixj];
          bool sw = desc ? (a < d) : (a > d);
          if (sw) { keys[i] = d; keys[ixj] = a; }
        }
      }
      __syncthreads();
    }
  }
  for (int i = tid; i < TOPK; i += 1024) {
    unsigned idx = ~((unsigned)(keys[i] & 0xFFFFFFFFull));
    if (idx >= DKD_N) idx = 0;
    kpidx[b * TOPK + i] = (int)idx;
  }
}

// ---------------- keypoint refinement (WMMA) ----------------
// B-matrix column values: n=0 -> gx, n=1 -> gy, n=2 -> 1, n=3 -> gx^2+gy^2
__device__ __forceinline__ float dkd_bval(int k, int n) {
  if (k >= 25) return 0.0f;
  float gx = (float)(k % 5) - 2.0f;
  float gy = (float)(k / 5) - 2.0f;
  if (n == 0) return gx;
  if (n == 1) return gy;
  if (n == 2) return 1.0f;
  if (n == 3) return gx * gx + gy * gy;
  return 0.0f;
}

__global__ void dkd_kpts(const float* __restrict__ s,
                         const int* __restrict__ kpidx,
                         float* __restrict__ out) {
  __shared__ float eLds[4][28][16];   // [wave][tap][kpt] exp values (taps 25..27 = 0)
  __shared__ float bLds[28][16];      // block-invariant B matrix [tap][moment-col]
  __shared__ float dLds[4][4][16];    // [wave][moment][kpt]
  int tid = threadIdx.x;
  int w = tid >> 5;            // wave in block
  int l = tid & 31;            // lane (wave32)
  int h = l >> 4;              // lane half
  int m = l & 15;              // kpt-in-wave / column index
  int b = blockIdx.y;
  int kq = blockIdx.x * 64 + w * 16 + m;

  // fill constant B matrix once per block (branchy select runs once, not per WMMA)
  for (int i = tid; i < 28 * 16; i += 128) {
    bLds[i >> 4][i & 15] = dkd_bval(i >> 4, i & 15);
  }

  int idx = kpidx[b * TOPK + kq];
  if (idx < 0 || idx >= DKD_N) idx = 0;
  int iy = idx >> 10;
  int ix = idx & (DKD_W - 1);
  if (iy < 2) iy = 2; if (iy > DKD_H - 3) iy = DKD_H - 3;
  if (ix < 2) ix = 2; if (ix > DKD_W - 3) ix = DKD_W - 3;
  const float* sb = s + (size_t)b * DKD_N;

  // pass 1: raw patch values into LDS + running max over own taps
  int t0 = h ? 13 : 0;
  int t1 = h ? 25 : 13;
  float mx = -1e30f;
  for (int t = t0; t < t1; ++t) {
    int dy = t / 5, dx = t % 5;
    float v = sb[(iy + dy - 2) * DKD_W + (ix + dx - 2)];
    eLds[w][t][m] = v;
    mx = fmaxf(mx, v);
  }
  if (h) { eLds[w][25][m] = 0.0f; eLds[w][26][m] = 0.0f; eLds[w][27][m] = 0.0f; }
  mx = fmaxf(mx, __shfl_xor(mx, 16));

  // pass 2: exp((v - max) / 0.1) in place
  for (int t = t0; t < t1; ++t) {
    float v = eLds[w][t][m];
    eLds[w][t][m] = __expf((v - mx) * 10.0f);
  }
  __syncthreads();

  float Sx, Sy, Se, Sg;
#if __has_builtin(__builtin_amdgcn_wmma_f32_16x16x4_f32)
  // A: 16x4 f32 (M=kpt, K=tap chunk); lane l holds A[M=m][K=4c+2h +{0,1}]
  // B: 4x16 f32 (K=tap, N=moment);    lane l holds B[K=4c+2h +{0,1}][N=m]
  v8f acc = {};
#pragma unroll
  for (int c = 0; c < 7; ++c) {
    int k0 = 4 * c + 2 * h;
    v2f a, bv;
    a.x = eLds[w][k0][m];
    a.y = eLds[w][k0 + 1][m];
    bv.x = bLds[k0][m];
    bv.y = bLds[k0 + 1][m];
    acc = __builtin_amdgcn_wmma_f32_16x16x4_f32(
        false, a, false, bv, (short)0, acc, false, false);
  }
  // D layout: value (M,N) in lane N (+16 if M>=8), VGPR M&7 -> scatter cols 0..3
  if (m < 4) {
#pragma unroll
    for (int r = 0; r < 8; ++r) dLds[w][m][r + 8 * h] = acc[r];
  }
  __syncthreads();
  Sx = dLds[w][0][m];
  Sy = dLds[w][1][m];
  Se = dLds[w][2][m];
  Sg = dLds[w][3][m];
#else
  Sx = 0.0f; Sy = 0.0f; Se = 0.0f; Sg = 0.0f;
  for (int t = 0; t < 25; ++t) {
    float e = eLds[w][t][m];
    float gx = (float)(t % 5) - 2.0f;
    float gy = (float)(t / 5) - 2.0f;
    Sx += e * gx; Sy += e * gy; Se += e; Sg += e * (gx * gx + gy * gy);
  }
#endif

  float rx = Sx / Se;
  float ry = Sy / Se;
  // sum(e*dist2)/Se with dist2 = ((g-r)/R)^2 summed -> closed form, R^2 = 4
  float disp = (Sg - (Sx * Sx + Sy * Sy) / Se) / (Se * 4.0f);

  float kx = (((float)ix + rx) / (float)(DKD_W - 1)) * 2.0f - 1.0f;
  float ky = (((float)iy + ry) / (float)(DKD_H - 1)) * 2.0f - 1.0f;

  // bilinear with zero padding
  float px = (kx + 1.0f) * 0.5f * (float)(DKD_W - 1);
  float py = (ky + 1.0f) * 0.5f * (float)(DKD_H - 1);
  float x0f = floorf(px), y0f = floorf(py);
  float wx1 = px - x0f, wx0 = 1.0f - wx1;
  float wy1 = py - y0f, wy0 = 1.0f - wy1;
  float score = 0.0f;
#pragma unroll
  for (int t = 0; t < 4; ++t) {
    float xi = x0f + (float)(t & 1);
    float yi = y0f + (float)(t >> 1);
    float wgt = ((t & 1) ? wx1 : wx0) * ((t >> 1) ? wy1 : wy0);
    bool valid = (xi >= 0.0f) && (xi < (float)DKD_W) &&
                 (yi >= 0.0f) && (yi < (float)DKD_H);
    int xc = (int)xi; if (xc < 0) xc = 0; if (xc > DKD_W - 1) xc = DKD_W - 1;
    int yc = (int)yi; if (yc < 0) yc = 0; if (yc > DKD_H - 1) yc = DKD_H - 1;
    score += valid ? sb[yc * DKD_W + xc] * wgt : 0.0f;
  }

  if (h == 0) {
    int g = b * TOPK + kq;
    out[2 * g + 0] = kx;
    out[2 * g + 1] = ky;
    out[DKD_B * TOPK * 2 + g] = score;
    out[DKD_B * TOPK * 2 + DKD_B * TOPK + g] = disp;
  }
}

// ---------------- launch ----------------
extern "C" void kernel_launch(void* const* d_in, const int* in_sizes, int n_in,
                              void* d_out, int out_size, void* d_ws, size_t ws_size,
                              hipStream_t stream) {
  (void)in_sizes; (void)n_in; (void)out_size; (void)ws_size;
  const float* s = (const float*)d_in[0];
  float* out = (float*)d_out;
  unsigned char* wsb = (unsigned char*)d_ws;

  float* A = (float*)(wsb + OFF_A);
  unsigned char* maskb = wsb + OFF_MASK;
  unsigned char* tmpb  = wsb + OFF_TMPB;
  unsigned char* suppb = wsb + OFF_SUPP;
  unsigned* hist = (unsigned*)(wsb + OFF_HIST);
  unsigned* thr  = (unsigned*)(wsb + OFF_THR);
  unsigned* cnt  = (unsigned*)(wsb + OFF_CNT);
  unsigned long long* cand = (unsigned long long*)(wsb + OFF_CAND);
  int* kpidx = (int*)(wsb + OFF_KIDX);

  const int EB = (DKD_TOT + 255) / 256;  // elementwise blocks

  // init mask: s == maxpool5(s)
  dkd_rowmax<<<EB, 256, 0, stream>>>(s, suppb, A, 0);
  dkd_colcmp<<<EB, 256, 0, stream>>>(s, A, suppb, maskb, 0);

  // two NMS suppression iterations
  for (int it = 0; it < 2; ++it) {
    dkd_rowor<<<EB, 256, 0, stream>>>(maskb, tmpb);
    dkd_color<<<EB, 256, 0, stream>>>(tmpb, suppb);
    dkd_rowmax<<<EB, 256, 0, stream>>>(s, suppb, A, 1);
    dkd_colcmp<<<EB, 256, 0, stream>>>(s, A, suppb, maskb, 1);
  }

  // top-k selection
  const int ZB = (DKD_B * NBINS + DKD_B + 255) / 256;
  dkd_zero<<<ZB, 256, 0, stream>>>(hist, cnt);
  dkd_hist<<<EB, 256, 0, stream>>>(s, maskb, hist);
  dkd_thresh<<<DKD_B, 1024, 0, stream>>>(hist, thr);
  dkd_compact<<<EB, 256, 0, stream>>>(s, maskb, thr, cnt, cand);
  dkd_sort<<<DKD_B, 1024, SORT_N * sizeof(unsigned long long), stream>>>(cand, cnt, kpidx);

  // keypoint refinement: 64 kpts / block (4 waves x 16), 8 batches
  dkd_kpts<<<dim3(TOPK / 64, DKD_B), 128, 0, stream>>>(s, kpidx, out);
}